// NequipDecoder_74156905333529
// MI455X (gfx1250) — compile-verified
//
#include <hip/hip_runtime.h>
#include <hip/hip_bf16.h>
#include <math.h>

// ---------------------------------------------------------------------------
// NequIP decoder for MI455X (gfx1250, wave32).
// All dense matmuls go through V_WMMA_F32_16X16X4_F32 (fp32 in/out, exact
// match to the fp32 reference). Scatter-sum uses global_atomic_add_f32 (L2
// atomics; ~5.6M distinct accumulators -> negligible contention on 192MB L2).
// Edge-dim is processed in 8 chunks of 16000 so the per-edge radial-MLP
// output buffer stays ~20MB and the full workspace (~88MB) is L2-resident.
// GEMM is specialized (template) on contiguous-A so the A fragment loads
// lower to global_load_b64 in the hot per-edge GEMMs.
//
// d_in pointer mapping assumes jax.tree_util.tree_leaves(setup_inputs())
// order (dict keys sorted alphabetically at every nesting level):
//  0 distance_vectors [E,3] f32      1 edge_index [2,E] i32
//  2 params.embed [100,32]
//  3..5 params.fc W0[8,64] W1[64,64] W2[64,160]
//  layer0 @6 : lin1{0e} lin2{0e,1o} radial{W0,W1,W2} sc{0e}            (7 leaves)
//  layer1 @13: lin1{0e,0o,1e,1o} lin2{0e,0o,1e,1o} radial{x3} sc{x4}   (15 leaves)
//  layer2 @28: same as layer1                                           (15 leaves)
//  43 params.lin_out [64,32]   44 params.ln_beta [256]   45 params.ln_gamma [256]
//  46 z [8000] i32
// ---------------------------------------------------------------------------

typedef __attribute__((ext_vector_type(2))) float v2f;
typedef __attribute__((ext_vector_type(8))) float v8f;

constexpr int NN = 8000;      // nodes
constexpr int NE = 128000;    // edges
constexpr int CH = 16000;     // edge chunk (NE/CH = 8, multiple of 16)

// ---- workspace layout (floats) --------------------------------------------
constexpr long OFF_SH1   = 0;                          // NE*3
constexpr long OFF_RBF   = OFF_SH1   + (long)NE*3;     // NE*8
constexpr long OFF_H0E   = OFF_RBF   + (long)NE*8;     // NN*32
constexpr long OFF_H1E   = OFF_H0E   + (long)NN*32;    // NN*96  [n][c][d]
constexpr long OFF_H0O   = OFF_H1E   + (long)NN*96;    // NN*32
constexpr long OFF_H1O   = OFF_H0O   + (long)NN*32;    // NN*96
constexpr long OFF_Y0E   = OFF_H1O   + (long)NN*96;    // NN*32
constexpr long OFF_Y1E   = OFF_Y0E   + (long)NN*32;    // NN*96
constexpr long OFF_Y0O   = OFF_Y1E   + (long)NN*96;    // NN*32
constexpr long OFF_Y1O   = OFF_Y0O   + (long)NN*32;    // NN*96
constexpr long OFF_AGG0E = OFF_Y1O   + (long)NN*96;    // NN*64  [n][slot*32+c]
constexpr long OFF_AGG0O = OFF_AGG0E + (long)NN*64;    // NN*64
constexpr long OFF_AGG1E = OFF_AGG0O + (long)NN*64;    // NN*288 [n][slot*32+c][d]
constexpr long OFF_AGG1O = OFF_AGG1E + (long)NN*288;   // NN*288
constexpr long OFF_OUT0E = OFF_AGG1O + (long)NN*288;   // NN*96
constexpr long OFF_OUT0O = OFF_OUT0E + (long)NN*96;    // NN*32
constexpr long OFF_OUT1E = OFF_OUT0O + (long)NN*32;    // NN*96
constexpr long OFF_OUT1O = OFF_OUT1E + (long)NN*96;    // NN*96
constexpr long OFF_HID1  = OFF_OUT1O + (long)NN*96;    // CH*64
constexpr long OFF_HID2  = OFF_HID1  + (long)CH*64;    // CH*64
constexpr long OFF_WED   = OFF_HID2  + (long)CH*64;    // CH*320 (radial / fc out)
constexpr long OFF_FEAT  = OFF_WED   + (long)CH*320;   // CH*64

static __device__ __forceinline__ float sspf(float x) {
  // shifted softplus: log(1+e^x) - log 2, numerically stable
  return fmaxf(x, 0.f) + log1pf(expf(-fabsf(x))) - 0.6931471805599453f;
}

// ---------------------------------------------------------------------------
// Generic strided fp32 WMMA GEMM: Y[m,n] = act(scale * sum_k X[m,k]*W[k,n])
// X element (m,k) at X[m*ldx + k*sx]; W dense row-major [K,ldw];
// Y element (m,n) at Y[m*ldy + n*sy]. One wave -> one 16x16 tile.
// Requires M%16==0, K%4==0, N%16==0.
// CONTIG=true: sx==1 and X rows 8B-aligned -> A fragment via one b64 load.
// ---------------------------------------------------------------------------
template <bool CONTIG>
__global__ __launch_bounds__(128) void gemm_f32_wmma(
    const float* __restrict__ X, int ldx, int sx,
    const float* __restrict__ W, int ldw,
    float* __restrict__ Y, int ldy, int sy,
    int M, int K, int N, float scale, int act)
{
  int lane = threadIdx.x;                 // 0..31
  int m0 = (blockIdx.x * blockDim.y + threadIdx.y) * 16;
  int n0 = blockIdx.y * 16;
  if (m0 >= M) return;
  int half = lane >> 4;                   // 0: K=k,k+1   1: K=k+2,k+3
  int l    = lane & 15;
  const float* xrow = X + (long)(m0 + l) * ldx;
  v8f acc = {0.f,0.f,0.f,0.f,0.f,0.f,0.f,0.f};
  for (int k = 0; k < K; k += 4) {
    int ka = k + half * 2;
    v2f a;
    if (CONTIG) {
      a = *(const v2f*)(xrow + ka);       // global_load_b64
    } else {
      a.x = xrow[(ka + 0) * sx];
      a.y = xrow[(ka + 1) * sx];
    }
    v2f b; b.x = W[(long)(ka + 0) * ldw + n0 + l];
           b.y = W[(long)(ka + 1) * ldw + n0 + l];
    acc = __builtin_amdgcn_wmma_f32_16x16x4_f32(
        false, a, false, b, (short)0, acc, false, false);
  }
#pragma unroll
  for (int r = 0; r < 8; ++r) {
    int row = m0 + r + half * 8;
    float v = acc[r] * scale;
    if (act) v = sspf(v);
    Y[(long)row * ldy + (n0 + l) * sy] = v;
  }
}

// ---------------------------------------------------------------------------
__global__ void fill_kernel(float* p, long n, float v) {
  long i = (long)blockIdx.x * blockDim.x + threadIdx.x;
  if (i < n) p[i] = v;
}

__global__ void embed_kernel(const float* __restrict__ emb,
                             const int* __restrict__ z,
                             float* __restrict__ h0e, int N) {
  int t = blockIdx.x * blockDim.x + threadIdx.x;
  if (t >= N * 32) return;
  int n = t >> 5, c = t & 31;
  h0e[t] = emb[(long)z[n] * 32 + c] * 0.1f;   // 1/sqrt(N_ELEMS=100)
}

__global__ void edge_geom_kernel(const float* __restrict__ dv,
                                 float* __restrict__ sh1,
                                 float* __restrict__ rbf, int E) {
  int e = blockIdx.x * blockDim.x + threadIdx.x;
  if (e >= E) return;
  float x = dv[e * 3], y = dv[e * 3 + 1], z = dv[e * 3 + 2];
  float r = sqrtf(x * x + y * y + z * z);
  float inv = 1.f / r;
  const float S3 = 1.7320508075688772f;
  sh1[e * 3]     = S3 * x * inv;
  sh1[e * 3 + 1] = S3 * y * inv;
  sh1[e * 3 + 2] = S3 * z * inv;
  float xr = r * (1.f / 1.5f);
  float cut = 0.f;
  if (xr < 1.f) {
    float x2 = xr * xr, x3 = x2 * xr;
    float x6 = x3 * x3, x7 = x6 * xr, x8 = x7 * xr;
    cut = 1.f - 28.f * x6 + 48.f * x7 - 21.f * x8;  // p=6 poly cutoff
  }
  float a = 1.1547005383792515f * inv * cut;        // sqrt(2/1.5)/r * cutoff
  const float PIC = 3.14159265358979323846f / 1.5f;
#pragma unroll
  for (int n = 1; n <= 8; ++n)
    rbf[(long)e * 8 + n - 1] = a * sinf((float)n * PIC * r);
}

// ---- tensor-product message pass ------------------------------------------
struct PathD { short ik, ok, kind, slot; };  // irrep codes: 0=0e 1=0o 2=1e 3=1o
// kinds: 0=ss 1=sv 2=vs 3=dot 4=cross
__constant__ PathD cPaths1[2] = {{0,0,0,0},{0,3,1,0}};
__constant__ PathD cPathsF[10] = {
  {0,0,0,0},{0,3,1,0},{2,2,2,0},{2,1,3,0},{2,3,4,1},
  {1,1,0,1},{1,2,1,1},{3,3,2,2},{3,0,3,1},{3,2,4,2}};

__global__ __launch_bounds__(256) void message_kernel(
    const int* __restrict__ src, const int* __restrict__ dst,
    const float* __restrict__ sh1, const float* __restrict__ wbuf,
    const float* __restrict__ y0e, const float* __restrict__ y0o,
    const float* __restrict__ y1e, const float* __restrict__ y1o,
    float* agg0e, float* agg0o, float* agg1e, float* agg1o,
    int cs, int Ec, int P, int useF)
{
  long t = (long)blockIdx.x * blockDim.x + threadIdx.x;
  if (t >= (long)Ec * 32) return;
  int c  = (int)(t & 31);
  int el = (int)(t >> 5);
  int e  = cs + el;
  int s = src[e], d = dst[e];
  float shx = sh1[e * 3], shy = sh1[e * 3 + 1], shz = sh1[e * 3 + 2];
  const PathD* pp = useF ? cPathsF : cPaths1;
  const float* wrow = wbuf + (long)el * P * 32 + c;
  for (int p = 0; p < P; ++p) {
    PathD pd = pp[p];
    float wp = wrow[p * 32] * 0.2f;   // fold 1/sqrt(AVG_NEIGH=25)
    float xs = 0.f, vx = 0.f, vy = 0.f, vz = 0.f;
    if (pd.ik == 0)      xs = y0e[(long)s * 32 + c];
    else if (pd.ik == 1) xs = y0o[(long)s * 32 + c];
    else {
      const float* v = (pd.ik == 2 ? y1e : y1o) + ((long)s * 32 + c) * 3;
      vx = v[0]; vy = v[1]; vz = v[2];
    }
    switch (pd.kind) {
      case 0: {  // ss
        atomicAdd((pd.ok == 0 ? agg0e : agg0o) + (long)d * 64 + pd.slot * 32 + c,
                  wp * xs);
      } break;
      case 1: {  // sv
        float m = wp * xs;
        float* tg = (pd.ok == 2 ? agg1e : agg1o) + (long)d * 288 + (pd.slot * 32 + c) * 3;
        atomicAdd(tg,     m * shx);
        atomicAdd(tg + 1, m * shy);
        atomicAdd(tg + 2, m * shz);
      } break;
      case 2: {  // vs
        float* tg = (pd.ok == 2 ? agg1e : agg1o) + (long)d * 288 + (pd.slot * 32 + c) * 3;
        atomicAdd(tg,     wp * vx);
        atomicAdd(tg + 1, wp * vy);
        atomicAdd(tg + 2, wp * vz);
      } break;
      case 3: {  // dot / sqrt3
        float m = wp * (vx * shx + vy * shy + vz * shz) * 0.5773502691896258f;
        atomicAdd((pd.ok == 0 ? agg0e : agg0o) + (long)d * 64 + pd.slot * 32 + c, m);
      } break;
      case 4: {  // cross / sqrt2
        float k = wp * 0.7071067811865476f;
        float* tg = (pd.ok == 2 ? agg1e : agg1o) + (long)d * 288 + (pd.slot * 32 + c) * 3;
        atomicAdd(tg,     k * (vy * shz - vz * shy));
        atomicAdd(tg + 1, k * (vz * shx - vx * shz));
        atomicAdd(tg + 2, k * (vx * shy - vy * shx));
      } break;
    }
  }
}

// ---- self-connection: out[n,o(,d)] += sum_c h[n,c(,d)] * W[c, z[n], o] / sqrt(3200)
__global__ void sc_kernel(const float* __restrict__ h, const float* __restrict__ W,
                          const int* __restrict__ z, float* __restrict__ out,
                          int N, int O, int vec) {
  long t = (long)blockIdx.x * blockDim.x + threadIdx.x;
  if (t >= (long)N * O) return;
  int n = (int)(t / O), o = (int)(t % O);
  const float* Wz = W + (long)z[n] * O + o;   // W[c][z][o], step c -> 100*O
  long cstep = (long)100 * O;
  const float norm = 0.017677669529663688f;   // 1/sqrt(32*100)
  if (!vec) {
    float acc = 0.f;
    for (int cc = 0; cc < 32; ++cc)
      acc = fmaf(h[(long)n * 32 + cc], Wz[cc * cstep], acc);
    out[(long)n * O + o] += acc * norm;
  } else {
    float a0 = 0.f, a1 = 0.f, a2 = 0.f;
    for (int cc = 0; cc < 32; ++cc) {
      float w = Wz[cc * cstep];
      const float* hv = h + ((long)n * 32 + cc) * 3;
      a0 = fmaf(hv[0], w, a0); a1 = fmaf(hv[1], w, a1); a2 = fmaf(hv[2], w, a2);
    }
    float* ob = out + ((long)n * O + o) * 3;
    ob[0] += a0 * norm; ob[1] += a1 * norm; ob[2] += a2 * norm;
  }
}

// ---- gate nonlinearity -----------------------------------------------------
__global__ void gate_kernel(const float* __restrict__ o0e, const float* __restrict__ o0o,
                            const float* __restrict__ o1e, const float* __restrict__ o1o,
                            float* h0e, float* h0o, float* h1e, float* h1o, int N) {
  int t = blockIdx.x * blockDim.x + threadIdx.x;
  if (t >= N * 32) return;
  int n = t >> 5, c = t & 31;
  h0e[t] = sspf(o0e[(long)n * 96 + c]);
  float g1 = sspf(o0e[(long)n * 96 + 32 + c]);
  float g2 = sspf(o0e[(long)n * 96 + 64 + c]);
  h0o[t] = tanhf(o0o[t]);
#pragma unroll
  for (int d = 0; d < 3; ++d) {
    h1e[(long)t * 3 + d] = o1e[(long)t * 3 + d] * g1;
    h1o[(long)t * 3 + d] = o1o[(long)t * 3 + d] * g2;
  }
}

// ---- LayerNorm over 256 flat features, one wave32 per node -----------------
__global__ __launch_bounds__(256) void layernorm_kernel(
    float* h0e, float* h1e, float* h0o, float* h1o,
    const float* __restrict__ gamma, const float* __restrict__ beta, int N) {
  int node = blockIdx.x * blockDim.y + threadIdx.y;
  if (node >= N) return;
  int lane = threadIdx.x;
  float vals[8];
  float s = 0.f, ss = 0.f;
#pragma unroll
  for (int j = 0; j < 8; ++j) {
    int idx = lane * 8 + j;   // flat: [0e 0:32 | 1e 32:128 | 0o 128:160 | 1o 160:256]
    float v;
    if (idx < 32)       v = h0e[(long)node * 32 + idx];
    else if (idx < 128) v = h1e[(long)node * 96 + idx - 32];
    else if (idx < 160) v = h0o[(long)node * 32 + idx - 128];
    else                v = h1o[(long)node * 96 + idx - 160];
    vals[j] = v; s += v; ss += v * v;
  }
#pragma unroll
  for (int m = 16; m >= 1; m >>= 1) {
    s  += __shfl_xor(s,  m, 32);
    ss += __shfl_xor(ss, m, 32);
  }
  float mean = s * (1.f / 256.f);
  float var  = ss * (1.f / 256.f) - mean * mean;
  float rstd = rsqrtf(var + 1e-5f);
#pragma unroll
  for (int j = 0; j < 8; ++j) {
    int idx = lane * 8 + j;
    float v = (vals[j] - mean) * rstd * gamma[idx] + beta[idx];
    if (idx < 32)       h0e[(long)node * 32 + idx] = v;
    else if (idx < 128) h1e[(long)node * 96 + idx - 32] = v;
    else if (idx < 160) h0o[(long)node * 32 + idx - 128] = v;
    else                h1o[(long)node * 96 + idx - 160] = v;
  }
}

// ---- final TP: feat[e, 0:32]=w0*h0e[dst]; feat[e,32:64]=w4*dot(h1o[dst],sh1)/sqrt3
__global__ void feat_kernel(const int* __restrict__ dst, const float* __restrict__ sh1,
                            const float* __restrict__ wfc,
                            const float* __restrict__ h0e, const float* __restrict__ h1o,
                            float* __restrict__ feat, int cs, int Ec) {
  long t = (long)blockIdx.x * blockDim.x + threadIdx.x;
  if (t >= (long)Ec * 32) return;
  int c  = (int)(t & 31);
  int el = (int)(t >> 5);
  int e  = cs + el;
  int d  = dst[e];
  float w0 = wfc[(long)el * 160 + c];
  float w4 = wfc[(long)el * 160 + 128 + c];
  feat[(long)el * 64 + c] = w0 * h0e[(long)d * 32 + c];
  const float* v = h1o + ((long)d * 32 + c) * 3;
  float sdot = (v[0] * sh1[e * 3] + v[1] * sh1[e * 3 + 1] + v[2] * sh1[e * 3 + 2])
               * 0.5773502691896258f;
  feat[(long)el * 64 + 32 + c] = w4 * sdot;
}

// ---------------------------------------------------------------------------
extern "C" void kernel_launch(void* const* d_in, const int* in_sizes, int n_in,
                              void* d_out, int out_size, void* d_ws, size_t ws_size,
                              hipStream_t stream) {
  (void)in_sizes; (void)n_in; (void)out_size; (void)ws_size;
  auto F = [&](int i) { return (const float*)d_in[i]; };

  const float* dv  = F(0);
  const int* ei    = (const int*)d_in[1];
  const int* src   = ei;            // edge_index[0] = j
  const int* dstI  = ei + NE;       // edge_index[1] = i
  const float* emb = F(2);
  const float* fc0 = F(3); const float* fc1 = F(4); const float* fc2 = F(5);
  const float* lin_out = F(43);
  const float* ln_beta = F(44); const float* ln_gamma = F(45);
  const int* z = (const int*)d_in[46];

  struct LP {
    const float *l1_0e,*l1_0o,*l1_1e,*l1_1o;
    const float *l2_0e,*l2_0o,*l2_1e,*l2_1o;
    const float *rw0,*rw1,*rw2;
    const float *sc0e,*sc0o,*sc1e,*sc1o;
    int P, k0e, k0o, k1e, k1o;
  } L[3];
  L[0] = {F(6), nullptr, nullptr, nullptr,
          F(7), nullptr, nullptr, F(8),
          F(9), F(10), F(11),
          F(12), nullptr, nullptr, nullptr,
          2, 1, 0, 0, 1};
  for (int li = 1, b = 13; li < 3; ++li, b += 15)
    L[li] = {F(b), F(b+1), F(b+2), F(b+3),
             F(b+4), F(b+5), F(b+6), F(b+7),
             F(b+8), F(b+9), F(b+10),
             F(b+11), F(b+12), F(b+13), F(b+14),
             10, 2, 2, 3, 3};

  float* ws = (float*)d_ws;
  float* sh1   = ws + OFF_SH1;   float* rbf   = ws + OFF_RBF;
  float* h0e   = ws + OFF_H0E;   float* h1e   = ws + OFF_H1E;
  float* h0o   = ws + OFF_H0O;   float* h1o   = ws + OFF_H1O;
  float* y0e   = ws + OFF_Y0E;   float* y1e   = ws + OFF_Y1E;
  float* y0o   = ws + OFF_Y0O;   float* y1o   = ws + OFF_Y1O;
  float* agg0e = ws + OFF_AGG0E; float* agg0o = ws + OFF_AGG0O;
  float* agg1e = ws + OFF_AGG1E; float* agg1o = ws + OFF_AGG1O;
  float* out0e = ws + OFF_OUT0E; float* out0o = ws + OFF_OUT0O;
  float* out1e = ws + OFF_OUT1E; float* out1o = ws + OFF_OUT1O;
  float* hid1  = ws + OFF_HID1;  float* hid2  = ws + OFF_HID2;
  float* wed   = ws + OFF_WED;   float* feat  = ws + OFF_FEAT;
  float* outp  = (float*)d_out;

  auto gemm = [&](const float* X, int ldx, int sx, const float* W, int ldw,
                  float* Y, int ldy, int sy, int M, int K, int Nc,
                  float scale, int act) {
    dim3 blk(32, 4, 1);
    dim3 grd((unsigned)((M / 16 + 3) / 4), (unsigned)(Nc / 16), 1);
    if (sx == 1)
      gemm_f32_wmma<true><<<grd, blk, 0, stream>>>(X, ldx, sx, W, ldw, Y, ldy, sy,
                                                   M, K, Nc, scale, act);
    else
      gemm_f32_wmma<false><<<grd, blk, 0, stream>>>(X, ldx, sx, W, ldw, Y, ldy, sy,
                                                    M, K, Nc, scale, act);
  };
  auto fill = [&](float* p, long n) {
    fill_kernel<<<(unsigned)((n + 255) / 256), 256, 0, stream>>>(p, n, 0.f);
  };

  const float S8  = 0.35355339059327373f;   // 1/sqrt(8)
  const float S32 = 0.17677669529663687f;   // 1/sqrt(32)
  const float S64 = 0.125f;                 // 1/sqrt(64)
  const float S96 = 0.10206207261596575f;   // 1/sqrt(96)

  // edge geometry + embedding
  edge_geom_kernel<<<(NE + 255) / 256, 256, 0, stream>>>(dv, sh1, rbf, NE);
  embed_kernel<<<(NN * 32 + 255) / 256, 256, 0, stream>>>(emb, z, h0e, NN);

  for (int li = 0; li < 3; ++li) {
    const LP& P = L[li];
    fill(agg0e, (long)NN * 704);   // agg0e..agg1o contiguous
    fill(out0e, (long)NN * 320);   // out0e..out1o contiguous

    // lin1 (per-irrep channel mixing), scale 1/sqrt(C)
    gemm(h0e, 32, 1, P.l1_0e, 32, y0e, 32, 1, NN, 32, 32, S32, 0);
    if (P.l1_0o) gemm(h0o, 32, 1, P.l1_0o, 32, y0o, 32, 1, NN, 32, 32, S32, 0);
    if (P.l1_1e) for (int d = 0; d < 3; ++d)
      gemm(h1e + d, 96, 3, P.l1_1e, 32, y1e + d, 96, 3, NN, 32, 32, S32, 0);
    if (P.l1_1o) for (int d = 0; d < 3; ++d)
      gemm(h1o + d, 96, 3, P.l1_1o, 32, y1o + d, 96, 3, NN, 32, 32, S32, 0);

    // per-edge radial MLP -> TP messages -> atomic scatter-sum (chunked)
    for (int cs = 0; cs < NE; cs += CH) {
      gemm(rbf + (long)cs * 8, 8, 1, P.rw0, 64, hid1, 64, 1, CH, 8, 64, S8, 1);
      gemm(hid1, 64, 1, P.rw1, 64, hid2, 64, 1, CH, 64, 64, S64, 1);
      gemm(hid2, 64, 1, P.rw2, P.P * 32, wed, P.P * 32, 1, CH, 64, P.P * 32, S64, 0);
      message_kernel<<<(unsigned)(((long)CH * 32 + 255) / 256), 256, 0, stream>>>(
          src, dstI, sh1, wed, y0e, y0o, y1e, y1o,
          agg0e, agg0o, agg1e, agg1o, cs, CH, P.P, li > 0 ? 1 : 0);
    }

    // lin2, scale 1/sqrt(k*C)
    gemm(agg0e, 64, 1, P.l2_0e, 96, out0e, 96, 1, NN, P.k0e * 32, 96,
         P.k0e == 1 ? S32 : S64, 0);
    if (P.l2_0o)
      gemm(agg0o, 64, 1, P.l2_0o, 32, out0o, 32, 1, NN, P.k0o * 32, 32, S64, 0);
    if (P.l2_1e) for (int d = 0; d < 3; ++d)
      gemm(agg1e + d, 288, 3, P.l2_1e, 32, out1e + d, 96, 3, NN, P.k1e * 32, 32, S96, 0);
    if (P.l2_1o) for (int d = 0; d < 3; ++d)
      gemm(agg1o + d, 288, 3, P.l2_1o, 32, out1o + d, 96, 3, NN, P.k1o * 32, 32,
           P.k1o == 1 ? S32 : S96, 0);

    // self connection
    sc_kernel<<<(unsigned)(((long)NN * 96 + 255) / 256), 256, 0, stream>>>(
        h0e, P.sc0e, z, out0e, NN, 96, 0);
    if (P.sc0o) sc_kernel<<<(unsigned)(((long)NN * 32 + 255) / 256), 256, 0, stream>>>(
        h0o, P.sc0o, z, out0o, NN, 32, 0);
    if (P.sc1e) sc_kernel<<<(unsigned)(((long)NN * 32 + 255) / 256), 256, 0, stream>>>(
        h1e, P.sc1e, z, out1e, NN, 32, 1);
    if (P.sc1o) sc_kernel<<<(unsigned)(((long)NN * 32 + 255) / 256), 256, 0, stream>>>(
        h1o, P.sc1o, z, out1o, NN, 32, 1);

    // gate -> new h
    gate_kernel<<<(NN * 32 + 255) / 256, 256, 0, stream>>>(
        out0e, out0o, out1e, out1o, h0e, h0o, h1e, h1o, NN);
  }

  // LayerNorm over 256 flat features per node
  {
    dim3 blk(32, 8, 1);
    layernorm_kernel<<<(NN + 7) / 8, blk, 0, stream>>>(
        h0e, h1e, h0o, h1o, ln_gamma, ln_beta, NN);
  }

  // final: fc MLP on rbf -> TP(0e blocks) -> lin_out (chunked over edges)
  for (int cs = 0; cs < NE; cs += CH) {
    gemm(rbf + (long)cs * 8, 8, 1, fc0, 64, hid1, 64, 1, CH, 8, 64, S8, 1);
    gemm(hid1, 64, 1, fc1, 64, hid2, 64, 1, CH, 64, 64, S64, 1);
    gemm(hid2, 64, 1, fc2, 160, wed, 160, 1, CH, 64, 160, S64, 0);
    feat_kernel<<<(unsigned)(((long)CH * 32 + 255) / 256), 256, 0, stream>>>(
        dstI, sh1, wed, h0e, h1o, feat, cs, CH);
    gemm(feat, 64, 1, lin_out, 32, outp + (long)cs * 32, 32, 1, CH, 64, 32, S64, 0);
  }
}